// BotRGCN34_32495722562037
// MI455X (gfx1250) — compile-verified
//
#include <hip/hip_runtime.h>

#define D 128

typedef __attribute__((ext_vector_type(2))) float v2f;
typedef __attribute__((ext_vector_type(8))) float v8f;

// ---------------------------------------------------------------------------
// Input embed: x[:, :64] = lrelu(num_prop @ W_np + b_np)
//              x[:, 64:] = lrelu(cat_prop @ W_cp + b_cp)
// One thread per (node, channel). Fan-in is tiny (6 / 11).
// ---------------------------------------------------------------------------
__global__ __launch_bounds__(256) void embed_kernel(
    const float* __restrict__ num_prop, const float* __restrict__ cat_prop,
    const float* __restrict__ W_np, const float* __restrict__ b_np,
    const float* __restrict__ W_cp, const float* __restrict__ b_cp,
    float* __restrict__ out, int N)
{
  int idx = blockIdx.x * blockDim.x + threadIdx.x;
  if (idx >= N * D) return;
  int i = idx >> 7;
  int c = idx & (D - 1);
  float s;
  if (c < 64) {
    s = b_np[c];
#pragma unroll
    for (int k = 0; k < 6; ++k) s += num_prop[i * 6 + k] * W_np[k * 64 + c];
  } else {
    int c2 = c - 64;
    s = b_cp[c2];
#pragma unroll
    for (int k = 0; k < 11; ++k) s += cat_prop[i * 11 + k] * W_cp[k * 64 + c2];
  }
  out[idx] = s > 0.0f ? s : 0.01f * s;
}

// ---------------------------------------------------------------------------
// Per-(relation,dst) in-degree counts. One thread per edge.
// ---------------------------------------------------------------------------
__global__ __launch_bounds__(256) void count_kernel(
    const int* __restrict__ edge_index, const int* __restrict__ edge_type,
    float* __restrict__ cnt, int E, int N)
{
  int e = blockIdx.x * blockDim.x + threadIdx.x;
  if (e >= E) return;
  int dst = edge_index[E + e];  // row 1 of [2,E]
  int et  = edge_type[e];
  atomicAdd(&cnt[(size_t)et * N + dst], 1.0f);
}

// ---------------------------------------------------------------------------
// Scatter-mean: out[dst] += xw[etype, src] / max(cnt[etype, dst], 1)
// One wave (32 lanes) per edge; each lane moves a float4 (32*4 = 128 = D).
// Equivalent to segment-mean per relation then sum over relations.
// ---------------------------------------------------------------------------
__global__ __launch_bounds__(256) void scatter_mean_kernel(
    const int* __restrict__ edge_index, const int* __restrict__ edge_type,
    const float* __restrict__ xw, const float* __restrict__ cnt,
    float* __restrict__ out, int E, int N)
{
  int gid  = blockIdx.x * blockDim.x + threadIdx.x;
  int e    = gid >> 5;
  if (e >= E) return;
  int lane = gid & 31;
  int src = edge_index[e];
  int dst = edge_index[E + e];
  int et  = edge_type[e];
  size_t seg = (size_t)et * N;
  float ic = 1.0f / fmaxf(cnt[seg + dst], 1.0f);
  const float4* sv = (const float4*)(xw + (seg + (size_t)src) * D);
  float4 v = sv[lane];
  float* dp = out + (size_t)dst * D + lane * 4;
  atomicAdd(dp + 0, v.x * ic);
  atomicAdd(dp + 1, v.y * ic);
  atomicAdd(dp + 2, v.z * ic);
  atomicAdd(dp + 3, v.w * ic);
}

// ---------------------------------------------------------------------------
// f32 WMMA GEMM: C[M,Ncols] (+)= A[M,K] @ W[K,Ncols] (+ bias) (lrelu?)
// One wave computes a 16x16 tile via K/4 chained V_WMMA_F32_16X16X4_F32.
// blockIdx.z batches over relations (W += z*strideW, C += z*strideC).
// A layout (ISA 7.12.2, 32-bit A 16x4):  M = lane&15, K = k0 + 2*(lane>>4) + {0,1}
// B layout (mirror):                     N = lane&15, K = k0 + 2*(lane>>4) + {0,1}
// C/D layout: vgpr v -> row m0 + v + 8*(lane>>4), col = n0 + (lane&15)
// Requires M % 16 == 0 (N = 100000 = 6250*16).
// ---------------------------------------------------------------------------
__global__ __launch_bounds__(256) void gemm_wmma_kernel(
    const float* __restrict__ A,
    const float* __restrict__ Wt,
    const float* __restrict__ bias,
    float* __restrict__ C,
    int M, int K, int Ncols, int ldc,
    int act, int accum,
    long long strideW, long long strideC)
{
  int wave = threadIdx.x >> 5;
  int lane = threadIdx.x & 31;
  int rowTile = blockIdx.y * (blockDim.x >> 5) + wave;
  if (rowTile * 16 >= M) return;  // wave-uniform: EXEC stays all-ones for WMMA

  Wt += (long long)blockIdx.z * strideW;
  C  += (long long)blockIdx.z * strideC;

  int m0   = rowTile << 4;
  int n0   = blockIdx.x << 4;
  int half = lane >> 4;     // 0: lanes 0-15, 1: lanes 16-31
  int l15  = lane & 15;
  int col  = n0 + l15;
  bool colOK = col < Ncols;
  int colc = colOK ? col : 0;  // clamp address, zero data below

  const float* Ap = A + (size_t)(m0 + l15) * K + 2 * half;

  v8f acc = {};
  for (int k0 = 0; k0 < K; k0 += 4) {
    v2f a;
    a.x = Ap[k0];
    a.y = Ap[k0 + 1];
    float b0 = Wt[(size_t)(k0 + 2 * half)     * Ncols + colc];
    float b1 = Wt[(size_t)(k0 + 2 * half + 1) * Ncols + colc];
    v2f b;
    b.x = colOK ? b0 : 0.0f;
    b.y = colOK ? b1 : 0.0f;
    acc = __builtin_amdgcn_wmma_f32_16x16x4_f32(
        /*neg_a=*/false, a, /*neg_b=*/false, b,
        /*c_mod=*/(short)0, acc, /*reuse_a=*/false, /*reuse_b=*/false);
  }

  if (colOK) {
    float bv = bias ? bias[col] : 0.0f;
#pragma unroll
    for (int v = 0; v < 8; ++v) {
      int r = m0 + v + 8 * half;
      size_t off = (size_t)r * ldc + col;
      float val = acc[v] + bv;
      if (accum) val += C[off];
      if (act) val = val > 0.0f ? val : 0.01f * val;
      C[off] = val;
    }
  }
}

// ---------------------------------------------------------------------------
extern "C" void kernel_launch(void* const* d_in, const int* in_sizes, int n_in,
                              void* d_out, int out_size, void* d_ws, size_t ws_size,
                              hipStream_t stream)
{
  // setup_inputs() order; des(0)/tweet(1) are unused by the reference.
  const float* num_prop = (const float*)d_in[2];
  const float* cat_prop = (const float*)d_in[3];
  const int*   edge_index = (const int*)d_in[4];
  const int*   edge_type  = (const int*)d_in[5];
  const float* W_np  = (const float*)d_in[6];
  const float* b_np  = (const float*)d_in[7];
  const float* W_cp  = (const float*)d_in[8];
  const float* b_cp  = (const float*)d_in[9];
  const float* W_in  = (const float*)d_in[10];
  const float* b_in  = (const float*)d_in[11];
  const float* W_rel = (const float*)d_in[12];
  const float* W_root= (const float*)d_in[13];
  const float* b_rg  = (const float*)d_in[14];
  const float* W_o1  = (const float*)d_in[15];
  const float* b_o1  = (const float*)d_in[16];
  const float* W_o2  = (const float*)d_in[17];
  const float* b_o2  = (const float*)d_in[18];

  int N = in_sizes[2] / 6;         // num_prop is [N,6]
  int E = in_sizes[5];             // edge_type is [E]
  int R = in_sizes[12] / (D * D);  // W_rel is [R,D,D]

  // Workspace: x_a[N*D] | x_b[N*D] | xw[R*N*D] | cnt[R*N]  (~360 MB)
  float* x_a = (float*)d_ws;
  float* x_b = x_a + (size_t)N * D;
  float* xw  = x_b + (size_t)N * D;
  float* cnt = xw + (size_t)R * N * D;

  auto launch_gemm = [&](const float* Ap, const float* Wp, const float* bp,
                         float* Cp, int M, int Nc, int act, int accum,
                         long long sW, long long sC, int batch) {
    dim3 grid((Nc + 15) / 16, ((M + 15) / 16 + 7) / 8, batch);
    gemm_wmma_kernel<<<grid, 256, 0, stream>>>(Ap, Wp, bp, Cp, M, D, Nc,
                                               Nc, act, accum, sW, sC);
  };

  // 1) embed -> x_b
  int tot = N * D;
  embed_kernel<<<(tot + 255) / 256, 256, 0, stream>>>(
      num_prop, cat_prop, W_np, b_np, W_cp, b_cp, x_b, N);

  // 2) x_a = lrelu(x_b @ W_in + b_in)
  launch_gemm(x_b, W_in, b_in, x_a, N, D, /*act*/1, /*accum*/0, 0, 0, 1);

  // 3) per-(relation,dst) counts (edges identical for both layers)
  hipMemsetAsync(cnt, 0, (size_t)R * N * sizeof(float), stream);
  count_kernel<<<(E + 255) / 256, 256, 0, stream>>>(edge_index, edge_type,
                                                    cnt, E, N);

  // 4) two RGCN layers, ping-pong x_a <-> x_b
  for (int layer = 0; layer < 2; ++layer) {
    const float* xin = (layer == 0) ? x_a : x_b;
    float*      xout = (layer == 0) ? x_b : x_a;

    // xw[r] = xin @ W_rel[r]   (batched over relations)
    launch_gemm(xin, W_rel, nullptr, xw, N, D, 0, 0,
                (long long)D * D, (long long)N * D, R);

    // xout = scatter-mean over edges
    hipMemsetAsync(xout, 0, (size_t)N * D * sizeof(float), stream);
    long long sthreads = (long long)E * 32;
    scatter_mean_kernel<<<(unsigned)((sthreads + 255) / 256), 256, 0, stream>>>(
        edge_index, edge_type, xw, cnt, xout, E, N);

    // xout += xin @ W_root + b_rgcn
    launch_gemm(xin, W_root, b_rg, xout, N, D, 0, /*accum*/1, 0, 0, 1);
  }
  // result of 2 layers is in x_a

  // 5) x_b = lrelu(x_a @ W_o1 + b_o1)
  launch_gemm(x_a, W_o1, b_o1, x_b, N, D, 1, 0, 0, 0, 1);

  // 6) d_out = x_b @ W_o2 + b_o2   ([N,2], column-masked tile)
  launch_gemm(x_b, W_o2, b_o2, (float*)d_out, N, 2, 0, 0, 0, 0, 1);
}